// GCN_72361609003220
// MI455X (gfx1250) — compile-verified
//
#include <hip/hip_runtime.h>
#include <hip/hip_bf16.h>
#include <math.h>

// ---------------------------------------------------------------------------
// GCN 2-layer forward for MI455X (gfx1250, wave32).
//   S1 = scatter-add over edges of val * X[col] into rows      (L2-resident)
//   H1 = tanh(S1 @ W1^T)   via V_WMMA_F32_16X16X4_F32          (fp32 accum)
//   S2 = scatter-add over edges of val * H1[col]
//   out= tanh(S2 @ W2^T)   via V_WMMA_F32_16X16X4_F32
// All intermediates (51MB each) fit in the 192MB L2, so SpMM gathers and
// float atomics stay on-chip. GEMM waves each own a 16-row strip across the
// full output width so the A operand is loaded once per k-step.
// ---------------------------------------------------------------------------

typedef __attribute__((ext_vector_type(2))) float v2f;
typedef __attribute__((ext_vector_type(8))) float v8f;

#define DIM_K 128  // inner dim for both layers

// Branch-free tanh: tanh(x) = sign(x) * (1-e)/(1+e), e = exp(-2|x|).
// ~1e-6 accurate, no EXEC-divergent slow path (unlike libm tanhf).
__device__ __forceinline__ float fast_tanh(float x) {
    float ax = fabsf(x);
    float e  = __expf(-2.0f * ax);                    // v_exp_f32 path
    float r  = (1.0f - e) * __builtin_amdgcn_rcpf(1.0f + e);
    return copysignf(r, x);
}

__global__ void gcn_zero_kernel(float* __restrict__ p, long long n) {
    long long i = (long long)blockIdx.x * blockDim.x + threadIdx.x;
    long long stride = (long long)gridDim.x * blockDim.x;
    for (; i < n; i += stride) p[i] = 0.0f;
}

// One wave per edge: 32 lanes x float4 = the full 128-wide feature row.
// Gather X[col] (one 512B coalesced read), scale by edge value, scatter-add
// into H[row] with 4 contiguous global_atomic_add_f32 per lane (L2-resident).
__global__ void gcn_spmm_kernel(const float* __restrict__ vals,
                                const int* __restrict__ row,
                                const int* __restrict__ col,
                                const float* __restrict__ X,
                                float* __restrict__ H,
                                int E) {
    int wavesPerBlock = blockDim.x >> 5;
    int e = blockIdx.x * wavesPerBlock + (threadIdx.x >> 5);
    int lane = threadIdx.x & 31;
    if (e >= E) return;
    float v = vals[e];
    int c = col[e];
    int r = row[e];
    const float4* src = (const float4*)(X + (long long)c * DIM_K);
    float4 m = src[lane];
    float* dst = H + (long long)r * DIM_K + lane * 4;
    atomicAdd(dst + 0, v * m.x);
    atomicAdd(dst + 1, v * m.y);
    atomicAdd(dst + 2, v * m.z);
    atomicAdd(dst + 3, v * m.w);
}

// Fused GEMM + tanh: O[m][n] = tanh( sum_k S[m][k] * W[n][k] ), K = 128.
// One wave computes a 16 x (NT*16) strip: NT chained-accumulator tiles,
// sharing a single A load per k-step (32 k-steps x NT v_wmma_f32_16x16x4_f32).
//
// VGPR layouts per CDNA5 ISA 7.12.2 (wave32):
//   A 16x4 f32 : lanes 0-15 -> M=lane,    K={0,1}; lanes 16-31 -> M=lane-16, K={2,3}
//   B  4x16 f32: lanes 0-15 -> N=lane,    K={0,1}; lanes 16-31 -> N=lane-16, K={2,3}
//   C/D 16x16  : VGPR r: lanes 0-15 -> M=r, N=lane; lanes 16-31 -> M=r+8, N=lane-16
template <int NT>
__global__ void gcn_gemm_tanh_kernel(const float* __restrict__ S,
                                     const float* __restrict__ W,
                                     float* __restrict__ O,
                                     int nNodes) {
    const int lane = threadIdx.x & 31;
    const int wavesPerBlock = blockDim.x >> 5;
    const int mTile = blockIdx.x * wavesPerBlock + (threadIdx.x >> 5);
    if (mTile * 16 >= nNodes) return;   // wave-uniform: EXEC stays all-ones

    const int nOut  = NT * 16;
    const int l15   = lane & 15;
    const int khalf = (lane < 16) ? 0 : 2;   // K sub-pair owned by this lane

    const float* aBase = S + (long long)(mTile * 16 + l15) * DIM_K + khalf;
    const float* bBase = W + (long long)l15 * DIM_K + khalf;

    v8f acc[NT];
#pragma unroll
    for (int t = 0; t < NT; ++t) acc[t] = (v8f){};

#pragma unroll 2
    for (int k = 0; k < DIM_K; k += 4) {
        v2f a;
        a.x = aBase[k];
        a.y = aBase[k + 1];
#pragma unroll
        for (int t = 0; t < NT; ++t) {
            const float* bp = bBase + (long long)t * 16 * DIM_K + k;
            v2f b;
            b.x = bp[0];
            b.y = bp[1];
            // emits v_wmma_f32_16x16x4_f32
            acc[t] = __builtin_amdgcn_wmma_f32_16x16x4_f32(
                /*neg_a=*/false, a, /*neg_b=*/false, b,
                /*c_mod=*/(short)0, acc[t], /*reuse_a=*/false, /*reuse_b=*/false);
        }
    }

    // Epilogue: branch-free tanh + store.
    const int mbase = mTile * 16 + ((lane < 16) ? 0 : 8);
#pragma unroll
    for (int t = 0; t < NT; ++t) {
        const int n = t * 16 + l15;
#pragma unroll
        for (int r = 0; r < 8; ++r) {
            O[(long long)(mbase + r) * nOut + n] = fast_tanh(acc[t][r]);
        }
    }
}

extern "C" void kernel_launch(void* const* d_in, const int* in_sizes, int n_in,
                              void* d_out, int out_size, void* d_ws, size_t ws_size,
                              hipStream_t stream) {
    const float* X        = (const float*)d_in[0];   // [N,128]
    const float* W1       = (const float*)d_in[1];   // [128,128]
    const float* W2       = (const float*)d_in[2];   // [64,128]
    const float* edgeVals = (const float*)d_in[3];   // [E]
    const int*   edgeRow  = (const int*)d_in[4];     // [E]
    const int*   edgeCol  = (const int*)d_in[5];     // [E]
    float* out = (float*)d_out;                       // [N,64]

    const int N = in_sizes[0] / DIM_K;   // 100000
    const int E = in_sizes[3];           // 1600000
    const long long featCount = (long long)N * DIM_K;

    float* S  = (float*)d_ws;            // SpMM accumulator (reused both layers)
    float* H1 = S + featCount;           // tanh(layer1) activations

    const int zeroBlocks = 2048, zeroThreads = 256;

    // Edge scatter: 8 edges (waves) per 256-thread block
    const int spmmThreads = 256;
    const int edgesPerBlock = spmmThreads / 32;
    const int spmmBlocks = (E + edgesPerBlock - 1) / edgesPerBlock;

    // GEMM: 1 wave per 16-row strip, 8 waves per block
    const int gemmThreads = 256;
    const int wavesPerBlock = gemmThreads / 32;
    const int mTiles = (N + 15) / 16;    // 6250
    const int gemmBlocks = (mTiles + wavesPerBlock - 1) / wavesPerBlock;

    // ---- layer 1 ----
    gcn_zero_kernel<<<zeroBlocks, zeroThreads, 0, stream>>>(S, featCount);
    gcn_spmm_kernel<<<spmmBlocks, spmmThreads, 0, stream>>>(
        edgeVals, edgeRow, edgeCol, X, S, E);
    gcn_gemm_tanh_kernel<8><<<gemmBlocks, gemmThreads, 0, stream>>>(
        S, W1, H1, N);                   // nOut = 128

    // ---- layer 2 ----
    gcn_zero_kernel<<<zeroBlocks, zeroThreads, 0, stream>>>(S, featCount);
    gcn_spmm_kernel<<<spmmBlocks, spmmThreads, 0, stream>>>(
        edgeVals, edgeRow, edgeCol, H1, S, E);
    gcn_gemm_tanh_kernel<4><<<gemmBlocks, gemmThreads, 0, stream>>>(
        S, W2, out, N);                  // nOut = 64
}